// SimpleGCN_77876347011160
// MI455X (gfx1250) — compile-verified
//
#include <hip/hip_runtime.h>

typedef float v2f __attribute__((ext_vector_type(2)));
typedef float v8f __attribute__((ext_vector_type(8)));

#define N_NODES  100000
#define N_EDGES  1600000
#define N_GRAPHS 512
#define DIM      128
#define DOUT     32

static __device__ __forceinline__ void atomic_add_f32(float* p, float v) {
  __hip_atomic_fetch_add(p, v, __ATOMIC_RELAXED, __HIP_MEMORY_SCOPE_AGENT);
}

// ---------------- degree / normalization ----------------
__global__ __launch_bounds__(256) void k_fill1(float* __restrict__ p, int n) {
  int i = blockIdx.x * 256 + threadIdx.x;
  if (i < n) p[i] = 1.0f;  // self-loop contributes 1 to every degree
}

__global__ __launch_bounds__(256) void k_zero(float* __restrict__ p, int n) {
  int i = blockIdx.x * 256 + threadIdx.x;
  if (i < n) p[i] = 0.0f;
}

__global__ __launch_bounds__(256) void k_deg_edges(const long long* __restrict__ dst,
                                                   float* __restrict__ deg, int E) {
  int e = blockIdx.x * 256 + threadIdx.x;
  if (e < E) atomic_add_f32(&deg[(int)dst[e]], 1.0f);
}

__global__ __launch_bounds__(256) void k_rsqrt(float* __restrict__ d, int n) {
  int i = blockIdx.x * 256 + threadIdx.x;
  if (i < n) d[i] = rsqrtf(d[i]);  // deg >= 1 always (self loop)
}

// ---------------- fp32 WMMA GEMM: Y[N,128] = X[N,128] @ W[128,128] ----------
// One 16x16 output tile per wave; 8 waves/block -> 16 rows x 128 cols per block.
// V_WMMA_F32_16X16X4_F32, K stepped by 4 (32 WMMAs per tile). Memory-bound, so
// fp32 matches the reference with no meaningful speed loss vs bf16.
__global__ __launch_bounds__(256) void k_gemm128_wmma(const float* __restrict__ X,
                                                      const float* __restrict__ W,
                                                      float* __restrict__ Y) {
  const int wave = threadIdx.x >> 5;
  const int lane = threadIdx.x & 31;
  const int hi = lane >> 4;      // half-wave select
  const int lo = lane & 15;
  const int m0 = blockIdx.x * 16;
  const int n0 = wave * 16;

  const float* __restrict__ xrow = X + (size_t)(m0 + lo) * DIM;
  v8f c = {};
#pragma unroll
  for (int k0 = 0; k0 < DIM; k0 += 4) {
    // A 16x4 fp32: VGPR0 holds K=k0 (lanes 0-15) / K=k0+2 (lanes 16-31), VGPR1 = +1
    const int ka = k0 + 2 * hi;
    v2f a;
    a.x = xrow[ka];
    a.y = xrow[ka + 1];
    // B 4x16 fp32: VGPR0 row K=k0+2*hi, VGPR1 row K=k0+2*hi+1, col striped on lanes
    v2f b;
    b.x = W[(size_t)(ka)     * DIM + n0 + lo];
    b.y = W[(size_t)(ka + 1) * DIM + n0 + lo];
    c = __builtin_amdgcn_wmma_f32_16x16x4_f32(false, a, false, b, (short)0, c,
                                              false, false);
  }
  // C/D: VGPR v -> row m0 + v + 8*hi, col n0 + lo
#pragma unroll
  for (int v = 0; v < 8; ++v) {
    Y[(size_t)(m0 + v + 8 * hi) * DIM + n0 + lo] = c[v];
  }
}

// ---------------- self-loop init: agg = xw * dinv^2 ----------------
__global__ __launch_bounds__(256) void k_selfloop(const float* __restrict__ xw,
                                                  const float* __restrict__ dinv,
                                                  float* __restrict__ agg) {
  int t = blockIdx.x * 256 + threadIdx.x;   // over N_NODES * 32 float4 chunks
  if (t >= N_NODES * 32) return;
  int node = t >> 5;
  float d = dinv[node];
  float s = d * d;
  float4 v = ((const float4*)xw)[t];
  v.x *= s; v.y *= s; v.z *= s; v.w *= s;
  ((float4*)agg)[t] = v;
}

// ---------------- edge scatter: agg[dst] += xw[src] * dinv[src]*dinv[dst] ----
// One wave per edge; lane L owns dims [4L, 4L+4). 512B contiguous gather per
// edge, 4 hardware f32 atomics per lane into the destination row.
__global__ __launch_bounds__(256) void k_scatter(const long long* __restrict__ src,
                                                 const long long* __restrict__ dst,
                                                 const float* __restrict__ dinv,
                                                 const float* __restrict__ xw,
                                                 float* __restrict__ agg, int E) {
  int e = (blockIdx.x * 256 + threadIdx.x) >> 5;
  int lane = threadIdx.x & 31;
  if (e >= E) return;
  int s = (int)src[e];
  int d = (int)dst[e];
  float w = dinv[s] * dinv[d];
  float4 v = ((const float4*)(xw + (size_t)s * DIM))[lane];
  float* out = agg + (size_t)d * DIM + lane * 4;
  atomic_add_f32(out + 0, v.x * w);
  atomic_add_f32(out + 1, v.y * w);
  atomic_add_f32(out + 2, v.z * w);
  atomic_add_f32(out + 3, v.w * w);
}

// ---------------- bias + relu (layer 1 epilogue, in place) ----------------
__global__ __launch_bounds__(256) void k_bias_relu(float* __restrict__ agg,
                                                   const float* __restrict__ b) {
  int t = blockIdx.x * 256 + threadIdx.x;   // N_NODES * 32 float4 chunks
  if (t >= N_NODES * 32) return;
  int q = t & 31;
  float4 v = ((float4*)agg)[t];
  float4 bb = ((const float4*)b)[q];
  v.x = fmaxf(v.x + bb.x, 0.0f);
  v.y = fmaxf(v.y + bb.y, 0.0f);
  v.z = fmaxf(v.z + bb.z, 0.0f);
  v.w = fmaxf(v.w + bb.w, 0.0f);
  ((float4*)agg)[t] = v;
}

// ---------------- pooling: segment-sum raw layer-2 aggregates ----------------
__global__ __launch_bounds__(256) void k_pool(const float* __restrict__ agg,
                                              const long long* __restrict__ batch,
                                              float* __restrict__ pool,
                                              float* __restrict__ cnt) {
  int node = (blockIdx.x * 256 + threadIdx.x) >> 5;
  int lane = threadIdx.x & 31;
  if (node >= N_NODES) return;
  int g = (int)batch[node];
  float4 v = ((const float4*)(agg + (size_t)node * DIM))[lane];
  float* p = pool + (size_t)g * DIM + lane * 4;
  atomic_add_f32(p + 0, v.x);
  atomic_add_f32(p + 1, v.y);
  atomic_add_f32(p + 2, v.z);
  atomic_add_f32(p + 3, v.w);
  if (lane == 0) atomic_add_f32(&cnt[g], 1.0f);
}

// ---------------- epilogue: out = relu(pool/cnt + b2) @ Wl + bl -------------
// mean(h + b2) == mean(h) + b2, so b2 folds in here.
__global__ __launch_bounds__(256) void k_final(const float* __restrict__ pool,
                                               const float* __restrict__ cnt,
                                               const float* __restrict__ b2,
                                               const float* __restrict__ Wl,
                                               const float* __restrict__ bl,
                                               float* __restrict__ out) {
  int t = blockIdx.x * 256 + threadIdx.x;
  if (t >= N_GRAPHS * DOUT) return;
  int g = t / DOUT;
  int c = t % DOUT;
  float inv = 1.0f / fmaxf(cnt[g], 1.0f);
  float acc = bl[c];
  const float* prow = pool + (size_t)g * DIM;
#pragma unroll 8
  for (int k = 0; k < DIM; ++k) {
    float gv = fmaxf(fmaf(prow[k], inv, b2[k]), 0.0f);
    acc = fmaf(gv, Wl[k * DOUT + c], acc);
  }
  out[t] = acc;
}

extern "C" void kernel_launch(void* const* d_in, const int* in_sizes, int n_in,
                              void* d_out, int out_size, void* d_ws, size_t ws_size,
                              hipStream_t stream) {
  const float*     x     = (const float*)d_in[0];
  const long long* eidx  = (const long long*)d_in[1];  // int64 [2, E]
  const long long* batch = (const long long*)d_in[2];  // int64 [N]
  const float*     W1    = (const float*)d_in[3];
  const float*     b1    = (const float*)d_in[4];
  const float*     W2    = (const float*)d_in[5];
  const float*     b2    = (const float*)d_in[6];
  const float*     Wl    = (const float*)d_in[7];
  const float*     bl    = (const float*)d_in[8];
  float*           out   = (float*)d_out;

  const int E = in_sizes[1] / 2;
  const long long* src = eidx;
  const long long* dst = eidx + E;

  // workspace carve-up (~103 MB)
  float* xw   = (float*)d_ws;            // N*128
  float* agg  = xw  + (size_t)N_NODES * DIM;  // N*128
  float* dinv = agg + (size_t)N_NODES * DIM;  // N
  float* pool = dinv + N_NODES;               // G*128
  float* cnt  = pool + (size_t)N_GRAPHS * DIM;// G

  const int B = 256;
  const int rowTiles   = N_NODES / 16;              // 6250 (exact)
  const int edgeBlocks = (E + 7) / 8;               // wave per edge
  const int vecBlocks  = (N_NODES * 32 + B - 1) / B;
  const int nodeWaveBlocks = (N_NODES + 7) / 8;     // wave per node

  // normalization: deg = 1 + sum_dst -> dinv = rsqrt(deg)
  k_fill1<<<(N_NODES + B - 1) / B, B, 0, stream>>>(dinv, N_NODES);
  k_deg_edges<<<(E + B - 1) / B, B, 0, stream>>>(dst, dinv, E);
  k_rsqrt<<<(N_NODES + B - 1) / B, B, 0, stream>>>(dinv, N_NODES);

  // zero pooling accumulators (must be per-call: graph replay)
  k_zero<<<(N_GRAPHS * DIM + N_GRAPHS + B - 1) / B, B, 0, stream>>>(pool,
                                                    N_GRAPHS * DIM + N_GRAPHS);

  // layer 1: xw = x @ W1 ; agg = norm-aggregate(xw) ; agg = relu(agg + b1)
  k_gemm128_wmma<<<rowTiles, B, 0, stream>>>(x, W1, xw);
  k_selfloop<<<vecBlocks, B, 0, stream>>>(xw, dinv, agg);
  k_scatter<<<edgeBlocks, B, 0, stream>>>(src, dst, dinv, xw, agg, E);
  k_bias_relu<<<vecBlocks, B, 0, stream>>>(agg, b1);

  // layer 2: xw = agg @ W2 ; agg = norm-aggregate(xw)   (b2 folded into epilogue)
  k_gemm128_wmma<<<rowTiles, B, 0, stream>>>(agg, W2, xw);
  k_selfloop<<<vecBlocks, B, 0, stream>>>(xw, dinv, agg);
  k_scatter<<<edgeBlocks, B, 0, stream>>>(src, dst, dinv, xw, agg, E);

  // pool + head
  k_pool<<<nodeWaveBlocks, B, 0, stream>>>(agg, batch, pool, cnt);
  k_final<<<(N_GRAPHS * DOUT + B - 1) / B, B, 0, stream>>>(pool, cnt, b2, Wl, bl,
                                                           out);
}